// EAPM_62809601736848
// MI455X (gfx1250) — compile-verified
//
#include <hip/hip_runtime.h>

// ---------------------------------------------------------------------------
// EAPM iterative projection for MI455X (gfx1250, wave32, WMMA).
// fp32 GEMMs via V_WMMA_F32_16X16X4_F32; B-tile staged with
// GLOBAL_LOAD_ASYNC_TO_LDS_B128 (ASYNCcnt); done-flag gated stream pipeline.
// ---------------------------------------------------------------------------

typedef __attribute__((ext_vector_type(2))) float v2f;
typedef __attribute__((ext_vector_type(4))) float v4f;
typedef __attribute__((ext_vector_type(8))) float v8f;

#define Bdim  2048
#define Ndim  2048
#define Mdim  1024
#define FREE  512
#define BM    128
#define BN    128
#define BK    16
#define LDSTR 20          // padded LDS row stride (floats): keeps 16B alignment, spreads banks
#define MAXIT 31          // MAX_ITER + 1 scan steps
#define TOL   1.0e-4f

// Low 32 bits of a flat shared-aperture pointer == wave-relative LDS byte offset.
__device__ __forceinline__ unsigned lds_off(const void* p)
{
    return (unsigned)(size_t)p;
}

// Async DMA: global -> LDS, 16 bytes per lane, tracked by ASYNCcnt.
__device__ __forceinline__ void async_ld_b128(unsigned ldsaddr, const void* gaddr)
{
    asm volatile("global_load_async_to_lds_b128 %0, %1, off"
                 :: "v"(ldsaddr), "v"(gaddr) : "memory");
}

__device__ __forceinline__ void wait_asynccnt0()
{
    asm volatile("s_wait_asynccnt 0" ::: "memory");
}

// ---------------------------------------------------------------------------
// Templated NT GEMM:  C[row, col] = sum_k Asrc[row, k] * Bmat[col, k]
//  MODE 0: out = bias + C                         (initial projection)
//  MODE 1: A gets relu on k>=FREE; r = bias + C - zref; store r; den[row] += sum r^2
//  MODE 2: e = C - bias(b_0); eqsum[col] += sum |e|   (convergence GEMM z @ A^T)
// ---------------------------------------------------------------------------
template <int MODE>
__global__ __launch_bounds__(256)
void k_gemm(const float* __restrict__ Asrc, const float* __restrict__ Bmat,
            const float* __restrict__ bias, const float* __restrict__ zref,
            float* __restrict__ outp, float* __restrict__ rowden,
            float* __restrict__ eqsum, const int* __restrict__ done,
            const int K, const int ldC)
{
    if (MODE != 0) {
        if (*done) return;              // uniform read -> uniform branch, EXEC stays full
    }

    __shared__ __align__(16) float As[2][BM * LDSTR];
    __shared__ __align__(16) float Bs[2][BN * LDSTR];

    const int tid   = threadIdx.x;
    const int lane  = tid & 31;
    const int half  = (lane >> 4) & 1;  // lane half selects K-pair / M+8 per ISA layout
    const int l15   = lane & 15;
    const int wid   = tid >> 5;         // 8 waves
    const int waveM = wid & 1;          // 2 waves over M (64 rows each)
    const int waveN = wid >> 1;         // 4 waves over N (32 cols each)
    const int rowBase = blockIdx.y * BM;
    const int colBase = blockIdx.x * BN;

    // global->LDS staging mapping: 256 threads x 2 rows x float4
    const int lrow = tid >> 2;          // 0..63
    const int lc4  = (tid & 3) << 2;    // 0,4,8,12

    v8f acc[4][2];
#pragma unroll
    for (int mi = 0; mi < 4; ++mi)
#pragma unroll
        for (int ni = 0; ni < 2; ++ni)
            acc[mi][ni] = (v8f){0.f, 0.f, 0.f, 0.f, 0.f, 0.f, 0.f, 0.f};

    v4f ra[2];

    // B tile: pure copy -> async DMA straight into LDS (no VGPR round-trip).
    auto asyncB = [&](int kt, int buf) {
        const int k0 = kt * BK;
#pragma unroll
        for (int i = 0; i < 2; ++i) {
            const int r = lrow + 64 * i;
            async_ld_b128(lds_off(&Bs[buf][r * LDSTR + lc4]),
                          Bmat + (size_t)(colBase + r) * K + (k0 + lc4));
        }
    };
    // A tile: needs the fused P2 relu (mode 1), so go through registers.
    auto gloadA = [&](int kt) {
        const int k0 = kt * BK;
#pragma unroll
        for (int i = 0; i < 2; ++i) {
            const int r = lrow + 64 * i;
            ra[i] = *(const v4f*)(Asrc + (size_t)(rowBase + r) * K + (k0 + lc4));
        }
        if (MODE == 1 && (k0 + lc4) >= FREE) {      // P2: relu on constrained columns
#pragma unroll
            for (int i = 0; i < 2; ++i)
#pragma unroll
                for (int e = 0; e < 4; ++e)
                    ra[i][e] = fmaxf(ra[i][e], 0.0f);
        }
    };
    auto storeA = [&](int buf) {
#pragma unroll
        for (int i = 0; i < 2; ++i) {
            const int r = lrow + 64 * i;
            *(v4f*)&As[buf][r * LDSTR + lc4] = ra[i];
        }
    };

    // Prologue: kick off DMA for tile 0 first, then stage A through regs.
    asyncB(0, 0);
    gloadA(0);
    storeA(0);
    wait_asynccnt0();                   // this wave's B DMA done before first barrier

    const int nK = K / BK;

    for (int kt = 0; kt < nK; ++kt) {
        const int buf = kt & 1;
        __syncthreads();                // all waves' stores + DMAs for `buf` visible
        const bool more = (kt + 1) < nK;
        if (more) {
            asyncB(kt + 1, buf ^ 1);    // start next-tile DMA in WMMA shadow
            gloadA(kt + 1);
        }

#pragma unroll
        for (int kc = 0; kc < 4; ++kc) {
            const int ko = kc * 4 + 2 * half;       // A: v0=K{0,2}, v1=K{1,3} per half
            v2f af[4], bf[2];
#pragma unroll
            for (int mi = 0; mi < 4; ++mi)
                af[mi] = *(const v2f*)&As[buf][(waveM * 64 + mi * 16 + l15) * LDSTR + ko];
#pragma unroll
            for (int ni = 0; ni < 2; ++ni)
                bf[ni] = *(const v2f*)&Bs[buf][(waveN * 32 + ni * 16 + l15) * LDSTR + ko];
#pragma unroll
            for (int mi = 0; mi < 4; ++mi)
#pragma unroll
                for (int ni = 0; ni < 2; ++ni)
                    acc[mi][ni] = __builtin_amdgcn_wmma_f32_16x16x4_f32(
                        false, af[mi], false, bf[ni], (short)0, acc[mi][ni], false, false);
        }

        if (more) {
            storeA(buf ^ 1);
            wait_asynccnt0();           // own DMA for buf^1 complete before next barrier
        }
    }

    const int wrow = rowBase + waveM * 64;
    const int wcol = colBase + waveN * 32;

    if (MODE == 0) {
#pragma unroll
        for (int mi = 0; mi < 4; ++mi)
#pragma unroll
            for (int i = 0; i < 8; ++i) {
                const int gr = wrow + mi * 16 + i + 8 * half;   // C/D: M = i + 8*half
#pragma unroll
                for (int ni = 0; ni < 2; ++ni) {
                    const int gc = wcol + ni * 16 + l15;
                    const size_t off = (size_t)gr * ldC + gc;
                    outp[off] = bias[off] + acc[mi][ni][i];
                }
            }
    } else if (MODE == 1) {
#pragma unroll
        for (int mi = 0; mi < 4; ++mi)
#pragma unroll
            for (int i = 0; i < 8; ++i) {
                const int gr = wrow + mi * 16 + i + 8 * half;
                float s = 0.0f;
#pragma unroll
                for (int ni = 0; ni < 2; ++ni) {
                    const int gc = wcol + ni * 16 + l15;
                    const size_t off = (size_t)gr * ldC + gc;
                    const float r = bias[off] + acc[mi][ni][i] - zref[off];
                    outp[off] = r;
                    s += r * r;
                }
                // reduce over the 16 lanes holding this row's columns
                s += __shfl_xor(s, 1, 32);
                s += __shfl_xor(s, 2, 32);
                s += __shfl_xor(s, 4, 32);
                s += __shfl_xor(s, 8, 32);
                if (l15 == 0) atomicAdd(&rowden[gr], s);
            }
    } else {  // MODE == 2: per-column sum of |C - b_0|
#pragma unroll
        for (int ni = 0; ni < 2; ++ni) {
            const int gc = wcol + ni * 16 + l15;
            float cs = 0.0f;
#pragma unroll
            for (int mi = 0; mi < 4; ++mi)
#pragma unroll
                for (int i = 0; i < 8; ++i) {
                    const int gr = wrow + mi * 16 + i + 8 * half;
                    cs += fabsf(acc[mi][ni][i] - bias[(size_t)gr * ldC + gc]);
                }
            atomicAdd(&eqsum[gc], cs);
        }
    }
}

// ---------------------------------------------------------------------------
// Per-row stats on current z: num[b] = sum_{j>=FREE} min(z,0)^2,
// feas[b] = all(z[b, FREE:] >= 0). Also zeroes per-iteration reduction scratch.
// ---------------------------------------------------------------------------
__global__ __launch_bounds__(256)
void k_rowstat(const float* __restrict__ z, float* __restrict__ num,
               float* __restrict__ feas, float* __restrict__ den,
               float* __restrict__ eqsum, float* __restrict__ ineqsum,
               const int* __restrict__ done)
{
    const int b = blockIdx.x;           // one block per row, 2048 blocks
    const int tid = threadIdx.x;
    if (tid == 0) {                     // zero this iteration's reduction scratch
        den[b] = 0.0f;
        ineqsum[b] = 0.0f;              // gridDim.x == Ndim
        if (b < Mdim) eqsum[b] = 0.0f;
    }
    if (*done) return;

    float s = 0.0f;
    int ok = 1;
    for (int j = FREE + tid; j < Ndim; j += 256) {
        const float v = z[(size_t)b * Ndim + j];
        const float nv = fminf(v, 0.0f);
        s += nv * nv;
        ok &= (v >= 0.0f) ? 1 : 0;
    }
    __shared__ float sred[256];
    __shared__ int   ored[256];
    sred[tid] = s;
    ored[tid] = ok;
    __syncthreads();
    for (int off = 128; off > 0; off >>= 1) {
        if (tid < off) { sred[tid] += sred[tid + off]; ored[tid] &= ored[tid + off]; }
        __syncthreads();
    }
    if (tid == 0) { num[b] = sred[0]; feas[b] = ored[0] ? 1.0f : 0.0f; }
}

// ---------------------------------------------------------------------------
// z += K[b] * residual ; accumulate per-column sum of relu(-z_new) for j>=FREE.
// Grid: (Ndim/256, Bdim/256); thread owns column j over a 256-row chunk.
// ---------------------------------------------------------------------------
__global__ __launch_bounds__(256)
void k_update(float* __restrict__ z, const float* __restrict__ residual,
              const float* __restrict__ num, const float* __restrict__ den,
              const float* __restrict__ feas, float* __restrict__ ineqsum,
              const int* __restrict__ done)
{
    if (*done) return;
    const int j  = blockIdx.x * 256 + threadIdx.x;
    const int r0 = blockIdx.y * 256;
    float isum = 0.0f;
    for (int b = r0; b < r0 + 256; ++b) {
        const float Kf = (feas[b] > 0.5f) ? 1.0f : (num[b] / den[b]);
        const size_t off = (size_t)b * Ndim + j;
        const float zn = z[off] + Kf * residual[off];   // RHO == 1
        z[off] = zn;
        if (j >= FREE) isum += fmaxf(-zn, 0.0f);
    }
    if (j >= FREE) atomicAdd(&ineqsum[j], isum);
}

// ---------------------------------------------------------------------------
// Convergence check: mean|z@A^T - b0| per col <= TOL and mean relu(-z) <= TOL.
// Sums compared against TOL * B. Sets done, counts iterations.
// ---------------------------------------------------------------------------
__global__ __launch_bounds__(256)
void k_check(const float* __restrict__ eqsum, const float* __restrict__ ineqsum,
             int* __restrict__ done, int* __restrict__ iters)
{
    if (*done) return;
    __shared__ int bad;
    if (threadIdx.x == 0) bad = 0;
    __syncthreads();
    const float thr = TOL * (float)Bdim;
    for (int idx = threadIdx.x; idx < Mdim + (Ndim - FREE); idx += 256) {
        const float v = (idx < Mdim) ? eqsum[idx] : ineqsum[FREE + (idx - Mdim)];
        if (!(v <= thr)) atomicOr(&bad, 1);     // NaN counts as not converged
    }
    __syncthreads();
    if (threadIdx.x == 0) {
        *iters = *iters + 1;
        if (bad == 0) *done = 1;
    }
}

__global__ void k_initflags(int* __restrict__ flags)
{
    if (threadIdx.x == 0) { flags[0] = 0; flags[1] = 0; }
}

__global__ void k_finalize(float* __restrict__ out, const int* __restrict__ iters)
{
    if (threadIdx.x == 0) out[(size_t)Bdim * Ndim] = (float)(*iters);
}

// ---------------------------------------------------------------------------
extern "C" void kernel_launch(void* const* d_in, const int* in_sizes, int n_in,
                              void* d_out, int out_size, void* d_ws, size_t ws_size,
                              hipStream_t stream)
{
    const float* z_in  = (const float*)d_in[0];   // [B, N]
    const float* BiasP = (const float*)d_in[1];   // [B, N]
    const float* b0    = (const float*)d_in[2];   // [B, M]
    const float* Amat  = (const float*)d_in[3];   // [M, N]
    const float* Wz    = (const float*)d_in[4];   // [N, N] (WzProj; W = WzProj^T)

    float* zcur = (float*)d_out;                  // live z state lives in d_out[0 : B*N)

    float* ws       = (float*)d_ws;
    float* residual = ws;                                    // B*N
    float* num      = residual + (size_t)Bdim * Ndim;        // B
    float* den      = num + Bdim;                            // B
    float* feas     = den + Bdim;                            // B
    float* eqsum    = feas + Bdim;                           // M
    float* ineqsum  = eqsum + Mdim;                          // N
    int*   flags    = (int*)(ineqsum + Ndim);                // [done, iters]
    int*   done     = flags;
    int*   iters    = flags + 1;

    const dim3 blk(256);
    const dim3 gGemmW(Ndim / BN, Bdim / BM);   // 16 x 16 blocks (z @ W)
    const dim3 gGemmA(Mdim / BN, Bdim / BM);   //  8 x 16 blocks (z @ A^T)

    k_initflags<<<1, 32, 0, stream>>>(flags);

    // Initial projection: z = Bias_Proj + z_in @ W
    k_gemm<0><<<gGemmW, blk, 0, stream>>>(z_in, Wz, BiasP, nullptr, zcur,
                                          nullptr, nullptr, nullptr, Ndim, Ndim);

    for (int it = 0; it < MAXIT; ++it) {
        k_rowstat<<<Bdim, blk, 0, stream>>>(zcur, num, feas, den, eqsum, ineqsum, done);
        k_gemm<1><<<gGemmW, blk, 0, stream>>>(zcur, Wz, BiasP, zcur, residual,
                                              den, nullptr, done, Ndim, Ndim);
        k_update<<<dim3(Ndim / 256, Bdim / 256), blk, 0, stream>>>(
            zcur, residual, num, den, feas, ineqsum, done);
        k_gemm<2><<<gGemmA, blk, 0, stream>>>(zcur, Amat, b0, nullptr, nullptr,
                                              nullptr, eqsum, done, Ndim, Mdim);
        k_check<<<1, blk, 0, stream>>>(eqsum, ineqsum, done, iters);
    }

    k_finalize<<<1, 32, 0, stream>>>(zcur, iters);
}